// CellEncoder_9466107920686
// MI455X (gfx1250) — compile-verified
//
#include <hip/hip_runtime.h>
#include <hip/hip_bf16.h>

// ---------------------------------------------------------------------------
// CellEncoder for MI455X (gfx1250, wave32):
//   1) segment-mean of gathered chunk features (sorted segment ids -> binary
//      search per cell, one wave per cell, float4 coalesced row loads)
//   2) [n_cells,128] @ W^T[128,256] + b via V_WMMA_F32_16X16X4_F32
// Memory-bound (~1 GB traffic @ 23.3 TB/s); f32 WMMA keeps full precision
// while exceeding the ~0.15 TFLOP/s the GEMM actually needs.
// ---------------------------------------------------------------------------

typedef __attribute__((ext_vector_type(2))) float v2f;
typedef __attribute__((ext_vector_type(4))) float v4f;
typedef __attribute__((ext_vector_type(8))) float v8f;

static constexpr int DIN  = 128;   // D_IN
static constexpr int DOUT = 256;   // D_OUT

// ---------------------------------------------------------------------------
// Kernel 1: per-cell segment mean.
// One wave32 per cell. Lanes split the 128-dim row into 32 x float4.
// ---------------------------------------------------------------------------
__device__ __forceinline__ int lower_bound_i32(const int* __restrict__ a, int n, int key) {
    int lo = 0, hi = n;
    while (lo < hi) {
        int mid = (lo + hi) >> 1;
        if (a[mid] < key) lo = mid + 1; else hi = mid;
    }
    return lo;
}

__global__ void __launch_bounds__(256)
cell_seg_mean_kernel(const float* __restrict__ feats,      // [N_CHUNKS, 128]
                     const int*   __restrict__ chunk_idx,  // [n_idx]
                     const int*   __restrict__ seg_ids,    // [n_idx], sorted
                     float*       __restrict__ cell_emb,   // [n_cells, 128] (ws)
                     int n_idx, int n_cells) {
    const int lane = threadIdx.x & 31;
    const int wave = threadIdx.x >> 5;
    const int cell = blockIdx.x * (blockDim.x >> 5) + wave;
    if (cell >= n_cells) return;

    // Sorted segment ids: contiguous [start, end) range for this cell.
    const int start = lower_bound_i32(seg_ids, n_idx, cell);
    const int end   = lower_bound_i32(seg_ids, n_idx, cell + 1);

    v4f acc = {0.0f, 0.0f, 0.0f, 0.0f};
    int j = start;
    // 2-deep manual unroll for a little more memory-level parallelism.
    for (; j + 2 <= end; j += 2) {
        const int i0 = chunk_idx[j];
        const int i1 = chunk_idx[j + 1];
        v4f r0 = *(const v4f*)(feats + (size_t)i0 * DIN + 4 * lane);
        v4f r1 = *(const v4f*)(feats + (size_t)i1 * DIN + 4 * lane);
        acc += r0;
        acc += r1;
    }
    for (; j < end; ++j) {
        const int i0 = chunk_idx[j];
        acc += *(const v4f*)(feats + (size_t)i0 * DIN + 4 * lane);
    }

    const int   count = end - start;
    const float inv   = 1.0f / (float)(count > 1 ? count : 1);
    acc *= inv;
    *(v4f*)(cell_emb + (size_t)cell * DIN + 4 * lane) = acc;
}

// ---------------------------------------------------------------------------
// Kernel 2: Y[n_cells,256] = X[n_cells,128] @ W^T + b using f32 WMMA.
// One wave per 16x16 output tile; K=128 -> 32 chained v_wmma_f32_16x16x4_f32.
//
// VGPR layouts (CDNA5 ISA 7.12.2, wave32):
//   A 16x4 f32 (2 VGPRs): lanes 0-15 hold row M=lane {K=0,K=1};
//                         lanes 16-31 hold row M=lane-16 {K=2,K=3}
//   B 4x16 f32 (2 VGPRs): VGPR v: lanes 0-15 -> (K=v,   N=lane),
//                                 lanes 16-31 -> (K=v+2, N=lane-16)
//   C/D 16x16 f32 (8 VGPRs): VGPR r: lanes 0-15 -> (M=r,   N=lane),
//                                    lanes 16-31 -> (M=r+8, N=lane-16)
// Both A and B per-lane data are a contiguous float2 at K-offset (hi?2:0).
// ---------------------------------------------------------------------------
__global__ void __launch_bounds__(256)
cell_proj_wmma_kernel(const float* __restrict__ X,    // [n_cells, 128] (ws)
                      const float* __restrict__ W,    // [256, 128] row-major
                      const float* __restrict__ bias, // [256]
                      float*       __restrict__ Y,    // [n_cells, 256]
                      int n_cells) {
    const int lane   = threadIdx.x & 31;
    const int wave   = threadIdx.x >> 5;
    const int waveId = blockIdx.x * (blockDim.x >> 5) + wave;

    const int ntilesN = DOUT / 16;                 // 16 tiles over 256 cols
    const int mtiles  = (n_cells + 15) >> 4;
    if (waveId >= mtiles * ntilesN) return;

    const int mt = waveId / ntilesN;
    const int nt = waveId - mt * ntilesN;
    const int m0 = mt << 4;
    const int n0 = nt << 4;

    const int  ln = lane & 15;
    const bool hi = lane >= 16;
    const int  kb = hi ? 2 : 0;                    // K sub-offset for this half

    // Clamp OOB rows for the A loads (EXEC must stay all-1s for WMMA);
    // results for clamped rows are discarded at the store.
    int arow = m0 + ln;
    if (arow >= n_cells) arow = n_cells - 1;

    const float* Aptr = X + (size_t)arow * DIN + kb;        // float2 @ +k
    const float* Bptr = W + (size_t)(n0 + ln) * DIN + kb;   // W row = out dim

    v8f acc = {0.f, 0.f, 0.f, 0.f, 0.f, 0.f, 0.f, 0.f};

#pragma unroll
    for (int k = 0; k < DIN; k += 4) {
        v2f a = *(const v2f*)(Aptr + k);
        v2f b = *(const v2f*)(Bptr + k);
        acc = __builtin_amdgcn_wmma_f32_16x16x4_f32(
            /*neg_a=*/false, a, /*neg_b=*/false, b,
            /*c_mod=*/(short)0, acc, /*reuse_a=*/false, /*reuse_b=*/false);
    }

    const int   col = n0 + ln;
    const float bv  = bias[col];
#pragma unroll
    for (int r = 0; r < 8; ++r) {
        const int row = m0 + r + (hi ? 8 : 0);
        if (row < n_cells) {
            Y[(size_t)row * DOUT + col] = acc[r] + bv;
        }
    }
}

// ---------------------------------------------------------------------------
// Host launcher
// in order: chunk_features[f32], W[f32], b[f32], chunk_idx[i32],
//           segment_ids[i32], n_cells[i32 scalar]
// ---------------------------------------------------------------------------
extern "C" void kernel_launch(void* const* d_in, const int* in_sizes, int n_in,
                              void* d_out, int out_size, void* d_ws, size_t ws_size,
                              hipStream_t stream) {
    const float* feats     = (const float*)d_in[0];
    const float* W         = (const float*)d_in[1];
    const float* bias      = (const float*)d_in[2];
    const int*   chunk_idx = (const int*)d_in[3];
    const int*   seg_ids   = (const int*)d_in[4];

    const int d_out_dim = in_sizes[2];            // 256
    const int n_idx     = in_sizes[3];            // 1.6M
    const int n_cells   = out_size / d_out_dim;   // 100K

    float* cell_emb = (float*)d_ws;               // [n_cells, 128] scratch

    // Kernel 1: 8 waves / block, one wave per cell.
    {
        const int waves_per_block = 256 / 32;
        const int blocks = (n_cells + waves_per_block - 1) / waves_per_block;
        cell_seg_mean_kernel<<<blocks, 256, 0, stream>>>(
            feats, chunk_idx, seg_ids, cell_emb, n_idx, n_cells);
    }

    // Kernel 2: one wave per 16x16 output tile.
    {
        const int ntilesN = d_out_dim / 16;
        const int mtiles  = (n_cells + 15) / 16;
        const int waves   = mtiles * ntilesN;
        const int waves_per_block = 256 / 32;
        const int blocks = (waves + waves_per_block - 1) / waves_per_block;
        cell_proj_wmma_kernel<<<blocks, 256, 0, stream>>>(
            cell_emb, W, bias, (float*)d_out, n_cells);
    }
}